// CRDLoss_14370960572549
// MI455X (gfx1250) — compile-verified
//
#include <hip/hip_runtime.h>
#include <hip/hip_bf16.h>
#include <math.h>

typedef __attribute__((ext_vector_type(2))) float v2f;
typedef __attribute__((ext_vector_type(8))) float v8f;

#define BATCH 128
#define FEAT  128
#define SDIM  2048
#define NCEK  4096
#define KTOT  4097            // 1 positive + 4096 negatives
#define NDATA 500000
#define NCET  0.07f

// Workspace layout (floats):
//  [0      , 16384)  : proj_s -> es (normalized in place)
//  [16384  , 32768)  : proj_t -> et
//  [32768  , +524416): out_v1  (exp(dot(mem2,es)/T))
//  [next   , +524416): out_v2  (exp(dot(mem1,et)/T))
//  [next   , +4)     : acc[0]=Zsum1 acc[1]=Zsum2 acc[2]=S1 acc[3]=S2
#define WS_ES   0
#define WS_ET   16384
#define WS_O1   32768
#define WS_O2   (WS_O1 + BATCH*KTOT)
#define WS_ACC  (WS_O2 + BATCH*KTOT)

__global__ void init_acc_kernel(float* acc) {
    if (threadIdx.x < 4) acc[threadIdx.x] = 0.0f;
}

// One wave (32 threads) computes one 16x16 fp32 tile of proj = A(128x2048) @ W(2048x128)
// using V_WMMA_F32_16X16X4_F32, K-loop of 512 steps.
// A 16x4 layout: lanes 0-15 hold (M=lane, K=0..1) in v0/v1, lanes 16-31 hold (M=lane-16, K=2..3).
// B 4x16 layout (symmetric): lanes 0-15 hold (K=0..1, N=lane), lanes 16-31 hold (K=2..3, N=lane-16).
// C 16x16 layout: VGPR r: lanes 0-15 -> (M=r, N=lane); lanes 16-31 -> (M=8+r, N=lane-16).
__global__ void proj_gemm_wmma(const float* __restrict__ fs, const float* __restrict__ ft,
                               const float* __restrict__ Ws, const float* __restrict__ Wt,
                               float* __restrict__ ws) {
    const int which = blockIdx.y;
    const float* __restrict__ A = which ? ft : fs;
    const float* __restrict__ W = which ? Wt : Ws;
    float* __restrict__ outp = ws + (which ? WS_ET : WS_ES);

    const int t    = blockIdx.x;          // 0..63
    const int m0   = (t >> 3) << 4;
    const int n0   = (t & 7) << 4;
    const int lane = threadIdx.x & 31;
    const int half = lane >> 4;           // 0: K=0,1  1: K=2,3
    const int l    = lane & 15;

    v8f c = {0.f,0.f,0.f,0.f,0.f,0.f,0.f,0.f};
    const float* __restrict__ arow = A + (size_t)(m0 + l) * SDIM;

    for (int k = 0; k < SDIM; k += 4) {
        v2f a, b;
        const float* ap = arow + k + 2 * half;       // contiguous pair -> b64 load
        a.x = ap[0];
        a.y = ap[1];
        const float* bp = W + (size_t)(k + 2 * half) * FEAT + n0 + l;
        b.x = bp[0];
        b.y = bp[FEAT];                               // next K row, same column
        c = __builtin_amdgcn_wmma_f32_16x16x4_f32(
                /*neg_a=*/false, a, /*neg_b=*/false, b,
                /*c_mod=*/(short)0, c, /*reuse_a=*/false, /*reuse_b=*/false);
    }

    const int rowbase = m0 + (half ? 8 : 0);
    const int col     = n0 + l;
#pragma unroll
    for (int r = 0; r < 8; ++r)
        outp[(size_t)(rowbase + r) * FEAT + col] = c[r];
}

// Per-row bias add + exact l2 normalization (matches reference: x / sqrt(sum(x*x))).
__global__ void rownorm_kernel(const float* __restrict__ bs, const float* __restrict__ bt,
                               float* __restrict__ ws) {
    const int which = blockIdx.y;
    const float* __restrict__ bias = which ? bt : bs;
    float* __restrict__ m = ws + (which ? WS_ET : WS_ES);
    const int b = blockIdx.x;
    const int n = threadIdx.x;      // 0..127

    __shared__ float sd[FEAT];
    float v = m[(size_t)b * FEAT + n] + bias[n];
    sd[n] = v * v;
    __syncthreads();
    for (int off = 64; off > 0; off >>= 1) {
        if (n < off) sd[n] += sd[n + off];
        __syncthreads();
    }
    m[(size_t)b * FEAT + n] = v / sqrtf(sd[0]);
}

// Bandwidth-dominant kernel: one wave per (b,k) row.
// 32 lanes x float4 = 512B coalesced gather of memory_v1/v2 rows, dot with
// register-resident e-vector chunks, shfl_xor wave reduction, exp, store +
// partial Z accumulation.
__global__ void gather_dot_exp(const int* __restrict__ idx, const int* __restrict__ cidx,
                               const float* __restrict__ mem1, const float* __restrict__ mem2,
                               float* __restrict__ ws, float* __restrict__ acc) {
    const float* __restrict__ ES = ws + WS_ES;
    const float* __restrict__ ET = ws + WS_ET;
    float* __restrict__ O1 = ws + WS_O1;
    float* __restrict__ O2 = ws + WS_O2;

    const int b     = blockIdx.y;
    const int lane  = threadIdx.x & 31;
    const int wave  = threadIdx.x >> 5;
    const int nwX   = gridDim.x * (blockDim.x >> 5);
    const int gw    = blockIdx.x * (blockDim.x >> 5) + wave;

    const float4 e_s = ((const float4*)(ES + (size_t)b * FEAT))[lane];
    const float4 e_t = ((const float4*)(ET + (size_t)b * FEAT))[lane];

    float z1 = 0.0f, z2 = 0.0f;
    for (int k = gw; k < KTOT; k += nwX) {
        const int id = (k == 0) ? idx[b] : cidx[(size_t)b * NCEK + (k - 1)];
        const float4 w1 = ((const float4*)(mem1 + (size_t)id * FEAT))[lane];
        const float4 w2 = ((const float4*)(mem2 + (size_t)id * FEAT))[lane];
        float d2 = w1.x * e_t.x + w1.y * e_t.y + w1.z * e_t.z + w1.w * e_t.w; // -> out_v2
        float d1 = w2.x * e_s.x + w2.y * e_s.y + w2.z * e_s.z + w2.w * e_s.w; // -> out_v1
#pragma unroll
        for (int off = 16; off > 0; off >>= 1) {
            d1 += __shfl_xor(d1, off);
            d2 += __shfl_xor(d2, off);
        }
        const float o1 = expf(d1 * (1.0f / NCET));
        const float o2 = expf(d2 * (1.0f / NCET));
        if (lane == 0) {
            O1[(size_t)b * KTOT + k] = o1;
            O2[(size_t)b * KTOT + k] = o2;
            z1 += o1;
            z2 += o2;
        }
    }

    __shared__ float s1[256], s2[256];
    const int tid = threadIdx.x;
    s1[tid] = z1;
    s2[tid] = z2;
    __syncthreads();
    for (int off = 128; off > 0; off >>= 1) {
        if (tid < off) { s1[tid] += s1[tid + off]; s2[tid] += s2[tid + off]; }
        __syncthreads();
    }
    if (tid == 0) {
        atomicAdd(&acc[0], s1[0]);
        atomicAdd(&acc[1], s2[0]);
    }
}

// Second pass: Z from accumulated sums, NCE log terms, block reduce + atomic.
__global__ void loss_kernel(const float* __restrict__ ws, float* __restrict__ acc) {
    const float* __restrict__ O1 = ws + WS_O1;
    const float* __restrict__ O2 = ws + WS_O2;
    const float Z1 = acc[0] * ((float)NDATA / (float)(BATCH * KTOT)); // mean * N_DATA
    const float Z2 = acc[1] * ((float)NDATA / (float)(BATCH * KTOT));
    const float mPn  = (float)NCEK / (float)NDATA;    // m * Pn
    const float cden = mPn + 1e-7f;                    // m*Pn + EPS

    float a1 = 0.0f, a2 = 0.0f;
    const int total = BATCH * KTOT;
    for (int i = blockIdx.x * blockDim.x + threadIdx.x; i < total;
         i += gridDim.x * blockDim.x) {
        const int k  = i % KTOT;
        const float o1 = O1[i] / Z1;
        const float o2 = O2[i] / Z2;
        a1 += (k == 0) ? logf(o1 / (o1 + cden)) : logf(mPn / (o1 + cden));
        a2 += (k == 0) ? logf(o2 / (o2 + cden)) : logf(mPn / (o2 + cden));
    }

    __shared__ float sa[256], sb[256];
    const int tid = threadIdx.x;
    sa[tid] = a1;
    sb[tid] = a2;
    __syncthreads();
    for (int off = 128; off > 0; off >>= 1) {
        if (tid < off) { sa[tid] += sa[tid + off]; sb[tid] += sb[tid + off]; }
        __syncthreads();
    }
    if (tid == 0) {
        atomicAdd(&acc[2], sa[0]);
        atomicAdd(&acc[3], sb[0]);
    }
}

__global__ void finalize_kernel(const float* __restrict__ acc, float* __restrict__ out) {
    // total = -(S1)/bsz - (S2)/bsz
    out[0] = -(acc[2] + acc[3]) * (1.0f / (float)BATCH);
}

extern "C" void kernel_launch(void* const* d_in, const int* in_sizes, int n_in,
                              void* d_out, int out_size, void* d_ws, size_t ws_size,
                              hipStream_t stream) {
    const float* f_s    = (const float*)d_in[0];
    const float* f_t    = (const float*)d_in[1];
    const int*   idx    = (const int*)  d_in[2];
    const int*   cidx   = (const int*)  d_in[3];
    const float* W_s    = (const float*)d_in[4];
    const float* b_s    = (const float*)d_in[5];
    const float* W_t    = (const float*)d_in[6];
    const float* b_t    = (const float*)d_in[7];
    const float* mem_v1 = (const float*)d_in[8];
    const float* mem_v2 = (const float*)d_in[9];

    float* ws  = (float*)d_ws;
    float* acc = ws + WS_ACC;
    float* out = (float*)d_out;

    init_acc_kernel<<<1, 32, 0, stream>>>(acc);

    // 64 tiles of 16x16 per matrix, 1 wave per tile, y selects s/t.
    proj_gemm_wmma<<<dim3(64, 2), 32, 0, stream>>>(f_s, f_t, W_s, W_t, ws);

    rownorm_kernel<<<dim3(BATCH, 2), FEAT, 0, stream>>>(b_s, b_t, ws);

    // 16 blocks x 8 waves = 128 waves per batch row; y = batch index.
    gather_dot_exp<<<dim3(16, BATCH), 256, 0, stream>>>(idx, cidx, mem_v1, mem_v2, ws, acc);

    loss_kernel<<<dim3(512), 256, 0, stream>>>(ws, acc);

    finalize_kernel<<<1, 1, 0, stream>>>(acc, out);
}